// Baseline_AnchorHeightPart_Single_27324581937310
// MI455X (gfx1250) — compile-verified
//
#include <hip/hip_runtime.h>

// CDNA5 / gfx1250: segmented mean+max pooling, bandwidth-bound (~252 MB @ 23.3 TB/s).
// Segment-sum:  one-hot x feats matmul via V_WMMA_F32_16X16X4_F32 (exact in fp32).
// Segment-max:  native float LDS atomic DS_MAX_NUM_F32 (atomicrmw fmax on shared).
// Streams feats with b128 loads; labels/mask staged in LDS, read back as b128.

typedef __attribute__((ext_vector_type(2))) float v2f;
typedef __attribute__((ext_vector_type(8))) float v8f;

#define N_ 32
#define C_ 128
#define S_ 30
#define K_ 512
#define P_ 16

__global__ void __launch_bounds__(256)
part_pool_kernel(const float* __restrict__ feats,
                 const int* __restrict__ labels,
                 const unsigned char* __restrict__ vmask,
                 float* __restrict__ out) {
  __shared__ alignas(16) int   lds_lab[K_];          // labels for this (n,s)
  __shared__ alignas(16) float lds_vm[K_];           // valid mask as float
  __shared__ alignas(16) float lds_max[8 * P_ * 16]; // per-wave [P][16 channels] running max
  __shared__ float lds_cnt[P_];                      // masked count per segment
  __shared__ int   lds_pcnt[P_];                     // raw patch count per segment

  const int blk = blockIdx.x;        // one workgroup per (n, s)
  const int n   = blk / S_;
  const int s   = blk % S_;
  const int tid = threadIdx.x;

  // ---- phase 1: stage labels + mask, init LDS accumulators ----
  const size_t nsbase = ((size_t)n * S_ + s) * (size_t)K_;
  for (int i = tid; i < K_; i += 256) {
    lds_lab[i] = labels[nsbase + i];
    lds_vm[i]  = (float)vmask[nsbase + i];
  }
  // init -100 implements reference's maximum(seg_max, -100) clamp
  for (int i = tid; i < 8 * P_ * 16; i += 256) lds_max[i] = -100.0f;
  if (tid < P_) { lds_cnt[tid] = 0.0f; lds_pcnt[tid] = 0; }
  __syncthreads();

  // ---- phase 2a: segment counts (one-shot LDS atomics) ----
  for (int i = tid; i < K_; i += 256) {
    const int lab = lds_lab[i];
    atomicAdd(&lds_cnt[lab], lds_vm[i]);      // ds_add_f32
    atomicAdd(&lds_pcnt[lab], 1);             // ds_add_u32
  }

  // ---- phase 2b: WMMA one-hot segment-sum + DS float scatter-max ----
  const int wave = tid >> 5;
  const int lane = tid & 31;
  const int half = lane >> 4;    // A/B layout: lanes 0-15 take low K pair, 16-31 high pair
  const int cidx = lane & 15;    // A-row p == B-col c == lane&15
  const int c    = wave * 16 + cidx;

  const float* fbase = feats + (((size_t)n * C_ + c) * S_ + s) * (size_t)K_;
  float* maxp = lds_max + wave * (P_ * 16);

  v8f acc = {0.f, 0.f, 0.f, 0.f, 0.f, 0.f, 0.f, 0.f};

  // Each iteration consumes 8 k-points: lane handles 4 contiguous floats
  // (k = 8j + 4*half .. +3) feeding TWO 16x16x4 WMMA chunks. Any permutation
  // of memory-k onto (chunk, K-slot) is valid for a segment sum.
  for (int j = 0; j < K_ / 8; ++j) {
    const int kk = j * 8 + half * 4;
    const float4 bv  = *(const float4*)(fbase + kk);     // global_load_b128
    const int4   lb  = *(const int4*)(lds_lab + kk);     // ds_load_b128
    const float4 vm4 = *(const float4*)(lds_vm + kk);    // ds_load_b128

    v2f a0, b0, a1, b1;
    a0.x = (lb.x == cidx) ? vm4.x : 0.0f;
    a0.y = (lb.y == cidx) ? vm4.y : 0.0f;
    b0.x = bv.x; b0.y = bv.y;
    a1.x = (lb.z == cidx) ? vm4.z : 0.0f;
    a1.y = (lb.w == cidx) ? vm4.w : 0.0f;
    b1.x = bv.z; b1.y = bv.w;

    acc = __builtin_amdgcn_wmma_f32_16x16x4_f32(
        false, a0, false, b0, (short)0, acc, false, false);
    acc = __builtin_amdgcn_wmma_f32_16x16x4_f32(
        false, a1, false, b1, (short)0, acc, false, false);

    // Unmasked segment-max via native DS_MAX_NUM_F32 (no monotone encoding).
    __hip_atomic_fetch_max(&maxp[lb.x * 16 + cidx], bv.x,
                           __ATOMIC_RELAXED, __HIP_MEMORY_SCOPE_WORKGROUP);
    __hip_atomic_fetch_max(&maxp[lb.y * 16 + cidx], bv.y,
                           __ATOMIC_RELAXED, __HIP_MEMORY_SCOPE_WORKGROUP);
    __hip_atomic_fetch_max(&maxp[lb.z * 16 + cidx], bv.z,
                           __ATOMIC_RELAXED, __HIP_MEMORY_SCOPE_WORKGROUP);
    __hip_atomic_fetch_max(&maxp[lb.w * 16 + cidx], bv.w,
                           __ATOMIC_RELAXED, __HIP_MEMORY_SCOPE_WORKGROUP);
  }
  __syncthreads();

  // ---- phase 3: finalize mean + max, write [n][c][s][P] ----
  // C/D layout: VGPR r holds row M = r + 8*half, col N = lane&15.
  const size_t outbase = (((size_t)n * C_ + c) * S_ + s) * (size_t)P_;
#pragma unroll
  for (int r = 0; r < 8; ++r) {
    const int p     = half * 8 + r;
    const float cnt = lds_cnt[p];
    const int   pc  = lds_pcnt[p];
    float mx = maxp[p * 16 + cidx];           // already >= -100 via init
    mx = (pc > 0) ? mx : 0.0f;                // empty segment -> 0
    const float mean = acc[r] / fmaxf(cnt, 1.0f);
    out[outbase + p] = mean + mx;             // p innermost: contiguous stores
  }
}

extern "C" void kernel_launch(void* const* d_in, const int* in_sizes, int n_in,
                              void* d_out, int out_size, void* d_ws, size_t ws_size,
                              hipStream_t stream) {
  (void)in_sizes; (void)n_in; (void)out_size; (void)d_ws; (void)ws_size;
  const float*         feats  = (const float*)d_in[0];
  const int*           labels = (const int*)d_in[1];
  const unsigned char* vmask  = (const unsigned char*)d_in[2];
  // d_in[3] = parts_num (== 16, baked into the WMMA tiling)
  float* out = (float*)d_out;

  part_pool_kernel<<<dim3(N_ * S_), dim3(256), 0, stream>>>(feats, labels, vmask, out);
}